// SingleChannelInterp_30623116820683
// MI455X (gfx1250) — compile-verified
//
#include <hip/hip_runtime.h>

typedef __attribute__((ext_vector_type(2))) float v2f;
typedef __attribute__((ext_vector_type(8))) float v8f;

#define REF_POINTS 192
#define D_INNER    12
#define NSRC       256

// One block per (batch, channel); 12 waves, one per 16-wide r-tile.
// Reduction over the 256 source points done with V_WMMA_F32_16X16X4_F32:
//   D(16x16) += A(16x4) x B(4x16)
//   A row 0 = ones      -> D row 0 = S[r]   (partition function)
//   A row 1 = x_t[k]    -> D row 1 = Y[r]   (weighted sum)
//   B[k,r]  = exp(-alpha*(d[k]-t_r)^2 + log m[k])   (computed on VALU)
__global__ __launch_bounds__(384)
void interp_wmma_kernel(const float* __restrict__ x,
                        const float* __restrict__ kern,
                        float* __restrict__ out)
{
    __shared__ float s_xt[NSRC];
    __shared__ float s_dd[NSRC];
    __shared__ float s_lm[NSRC];

    const int c    = blockIdx.x;      // channel 0..11
    const int b    = blockIdx.y;      // batch   0..63
    const int tid  = threadIdx.x;
    const int wave = tid >> 5;        // r-tile 0..11
    const int lane = tid & 31;
    const int col  = lane & 15;       // N index within tile
    const int kh   = lane >> 4;       // K half-select (wave32 layout)

    // Stage x_t, d, log(m) for this (b,c) into LDS (3 KB).
    const float* xb = x + ((size_t)b * 48 + c) * NSRC;
    if (tid < NSRC) {
        s_xt[tid] = xb[tid];
        s_lm[tid] = __logf(xb[D_INNER * NSRC + tid]);
        s_dd[tid] = xb[2 * D_INNER * NSRC + tid];
    }
    __syncthreads();

    const float alpha   = log1pf(__expf(kern[c]));   // softplus
    const float alpha10 = 10.0f * alpha;

    const int   r  = wave * 16 + col;
    const float tr = (float)r * (1.0f / 191.0f);     // linspace(0,48,192)/48

    v8f acc1  = {};   // alpha chain:  row0 = S, row1 = Y
    v8f acc10 = {};   // 10*alpha chain

    for (int k0 = 0; k0 < NSRC; k0 += 4) {
        const int k = k0 + 2 * kh;   // this half-wave's K pair

        const float d0 = s_dd[k], d1 = s_dd[k + 1];
        const float l0 = s_lm[k], l1 = s_lm[k + 1];
        const float x0 = s_xt[k], x1 = s_xt[k + 1];

        const float t0 = d0 - tr, t1 = d1 - tr;
        const float q0 = t0 * t0, q1 = t1 * t1;

        // B operand (4x16 f32): lanes 0-15 carry K=k0,k0+1; lanes 16-31 carry K=k0+2,k0+3
        v2f B1  = { __expf(fmaf(-alpha,   q0, l0)), __expf(fmaf(-alpha,   q1, l1)) };
        v2f B10 = { __expf(fmaf(-alpha10, q0, l0)), __expf(fmaf(-alpha10, q1, l1)) };

        // A operand (16x4 f32): row = col; row0 = ones, row1 = x_t, rest 0.
        const float a0 = (col == 0) ? 1.0f : ((col == 1) ? x0 : 0.0f);
        const float a1 = (col == 0) ? 1.0f : ((col == 1) ? x1 : 0.0f);
        v2f A = { a0, a1 };

        acc1  = __builtin_amdgcn_wmma_f32_16x16x4_f32(false, A, false, B1,
                                                      (short)0, acc1,  false, false);
        acc10 = __builtin_amdgcn_wmma_f32_16x16x4_f32(false, A, false, B10,
                                                      (short)0, acc10, false, false);
    }

    // D layout: VGPR v = row (v + 8*kh), lanes 0-15.  Rows 0/1 of each chain.
    if (kh == 0) {
        const float S1  = acc1[0]  + 1e-6f;
        const float Y1  = acc1[1];
        const float S10 = acc10[0] + 1e-6f;
        const float Y10 = acc10[1];

        float* ob = out + (size_t)b * 36 * REF_POINTS;
        ob[(c     ) * REF_POINTS + r] = Y1 / S1;      // y
        ob[(c + 12) * REF_POINTS + r] = __logf(S1);   // w = log(S + 1e-6)
        ob[(c + 24) * REF_POINTS + r] = Y10 / S10;    // y_trans
    }
}

extern "C" void kernel_launch(void* const* d_in, const int* in_sizes, int n_in,
                              void* d_out, int out_size, void* d_ws, size_t ws_size,
                              hipStream_t stream) {
    const float* x = (const float*)d_in[0];
    const float* k = (const float*)d_in[1];
    float* out     = (float*)d_out;

    dim3 grid(D_INNER, 64);   // (channel, batch) -> 768 blocks
    dim3 block(384);          // 12 wave32 waves, one per r-tile
    hipLaunchKernelGGL(interp_wmma_kernel, grid, block, 0, stream, x, k, out);
}